// Xor_56891136803062
// MI455X (gfx1250) — compile-verified
//
#include <hip/hip_runtime.h>
#include <hip/hip_bf16.h>

// ---------------------------------------------------------------------------
// Fully-fused 10-layer LSTM (H=4, T=512, B=4096) for gfx1250 (MI455X).
//
// One wave owns a 16-element batch tile; all weights/state live in VGPRs.
// Per layer-step, gate pre-activations come from two chained
// V_WMMA_F32_16X16X4_F32:  G(16 gates x 16 batch) = W_ih x in + W_hh x h + b.
//
// Two key tricks:
//  * GATE-ROW PERMUTATION: A-matrix rows are reordered to
//    (i0,i1,f0,f1,g0,g1,o0,o1 | i2,i3,f2,f3,g2,g3,o2,o3) so each lane-half
//    receives exactly the 8 gate values for the hidden channels it owns
//    (low lanes: j=0,1; high lanes: j=2,3). No cross-lane exchange is needed
//    after the WMMA, and h lands directly in next step's B-fragment layout.
//  * LAYER PIPELINING (wavefront skew): iteration n runs layer l at time
//    t = n - l. All 10 layer-steps per iteration depend only on previous-
//    iteration outputs -> 10 independent WMMA+activation chains for the
//    scheduler to overlap. Layers processed in descending order so no
//    snapshots are needed.
// ---------------------------------------------------------------------------

#define T_STEPS 512
#define B_TOTAL 4096
#define NLAYERS 10

typedef float v2f __attribute__((ext_vector_type(2)));
typedef float v8f __attribute__((ext_vector_type(8)));

__device__ __forceinline__ float half_swap(float v) {
    // ds_swizzle SWAPX16: dst[lane] = src[lane ^ 16]
    return __int_as_float(__builtin_amdgcn_ds_swizzle(__float_as_int(v), 0x401F));
}

#if __has_builtin(__builtin_amdgcn_tanhf)
__device__ __forceinline__ float fast_tanh(float x) {
    return __builtin_amdgcn_tanhf(x);               // v_tanh_f32
}
__device__ __forceinline__ float fast_sigmoid(float x) {
    return fmaf(0.5f, __builtin_amdgcn_tanhf(0.5f * x), 0.5f);
}
#else
__device__ __forceinline__ float fast_sigmoid(float x) {
    float e = __builtin_amdgcn_exp2f(-1.442695040888963f * x);
    return __builtin_amdgcn_rcpf(1.0f + e);
}
__device__ __forceinline__ float fast_tanh(float x) {
    float e = __builtin_amdgcn_exp2f(-2.885390081777927f * x);
    return 2.0f * __builtin_amdgcn_rcpf(1.0f + e) - 1.0f;
}
#endif

// Permuted source gate-row for logical D-row r (r = 0..15):
//   half = r>>3 selects hidden channels {0,1} vs {2,3};
//   idx  = r&7 enumerates (i,f,g,o) x (2 channels).
__device__ __forceinline__ int gate_row(int r) {
    int half = r >> 3, idx = r & 7;
    return (idx >> 1) * 4 + (half << 1) + (idx & 1);
}

__global__ __launch_bounds__(128, 1) void lstm10_fused_kernel(
    const float* __restrict__ x,          // (T, B, 1)
    const float* __restrict__ w_ih0,      // (16, 1)
    const float* __restrict__ w_ih_rest,  // (9, 16, 4)
    const float* __restrict__ w_hh,       // (10, 16, 4)
    const float* __restrict__ b_ih,       // (10, 16)
    const float* __restrict__ b_hh,       // (10, 16)
    const float* __restrict__ w_dec,      // (1, 4)
    const float* __restrict__ b_dec,      // (1,)
    float* __restrict__ out)              // (B, 1)
{
    const int lane  = threadIdx.x & 31;
    const int wave  = threadIdx.x >> 5;
    const int wpb   = blockDim.x >> 5;
    const int batch_base = (blockIdx.x * wpb + wave) * 16;
    const int m   = lane & 15;          // A: (permuted) gate row | B/D: batch col
    const bool hi = lane >= 16;         // high half-wave owns j = 2,3

    // ---- Resident A fragments (gate-row permuted) and bias C fragments -----
    // A 16x4 f32:  v0 = {K0 | K2} across lane halves, v1 = {K1 | K3}.
    // C/D 16x16 f32: vr = row r (lanes 0-15) / row r+8 (lanes 16-31).
    const int arow = gate_row(m);               // source row this lane loads
    v2f Aih[NLAYERS], Ahh[NLAYERS];
    v8f Cb[NLAYERS];
#pragma unroll
    for (int l = 0; l < NLAYERS; ++l) {
        const float* Whh = w_hh + l * 64;
        Ahh[l][0] = Whh[arow * 4 + (hi ? 2 : 0)];
        Ahh[l][1] = Whh[arow * 4 + (hi ? 3 : 1)];
        if (l == 0) {
            Aih[l][0] = hi ? 0.0f : w_ih0[arow];   // K=1 input, cols 1..3 = 0
            Aih[l][1] = 0.0f;
        } else {
            const float* Wih = w_ih_rest + (l - 1) * 64;
            Aih[l][0] = Wih[arow * 4 + (hi ? 2 : 0)];
            Aih[l][1] = Wih[arow * 4 + (hi ? 3 : 1)];
        }
#pragma unroll
        for (int r = 0; r < 8; ++r) {
            const int src = gate_row((hi ? 8 : 0) + r);
            Cb[l][r] = b_ih[l * 16 + src] + b_hh[l * 16 + src];
        }
    }

    // ---- Recurrent state, split across lane halves --------------------------
    // low lane b: h/c[b][0..1]   high lane b: h/c[b][2..3]  (= B-frag layout)
    float hst[NLAYERS][2];
    float cst[NLAYERS][2];
#pragma unroll
    for (int l = 0; l < NLAYERS; ++l) {
        hst[l][0] = 0.0f; hst[l][1] = 0.0f;
        cst[l][0] = 0.0f; cst[l][1] = 0.0f;
    }

    // One pipelined layer-step: layer l consumes Bin (= x frag or hst[l-1])
    auto layer_step = [&](int l, v2f Bin) {
        v2f Bh; Bh[0] = hst[l][0]; Bh[1] = hst[l][1];
        v8f acc = __builtin_amdgcn_wmma_f32_16x16x4_f32(
            false, Ahh[l], false, Bh, (short)0, Cb[l], false, false);
        acc = __builtin_amdgcn_wmma_f32_16x16x4_f32(
            false, Aih[l], false, Bin, (short)0, acc, false, false);
        // acc rows (this lane-half): 0,1 = i | 2,3 = f | 4,5 = g | 6,7 = o
#pragma unroll
        for (int k = 0; k < 2; ++k) {
            float cn = fast_sigmoid(acc[2 + k]) * cst[l][k] +
                       fast_sigmoid(acc[k])     * fast_tanh(acc[4 + k]);
            cst[l][k] = cn;
            hst[l][k] = fast_sigmoid(acc[6 + k]) * fast_tanh(cn);
        }
    };

    // One skewed iteration: layer l runs time t = n - l (descending l order).
    auto pipeline_iter = [&](int n, bool guard_lo, bool guard_hi) {
        v2f Bx; Bx[0] = 0.0f; Bx[1] = 0.0f;
        if (!guard_hi) {                     // layer 0 inactive in epilogue
            const float xl = x[n * B_TOTAL + batch_base + m];
            Bx[0] = hi ? 0.0f : xl;          // row K0 = x_t on low lanes
        }
#pragma unroll
        for (int li = 0; li < NLAYERS; ++li) {
            const int l = NLAYERS - 1 - li;
            if (guard_lo && n < l) continue;              // t < 0
            if (guard_hi && (n - l) >= T_STEPS) continue; // t >= T
            v2f Bin;
            if (l == 0) { Bin = Bx; }
            else        { Bin[0] = hst[l - 1][0]; Bin[1] = hst[l - 1][1]; }
            layer_step(l, Bin);
        }
    };

    // ---- Skewed time recurrence: n = t + l ----------------------------------
#pragma unroll 1
    for (int n = 0; n < NLAYERS - 1; ++n)            // ramp-up
        pipeline_iter(n, true, false);
#pragma unroll 1
    for (int n = NLAYERS - 1; n < T_STEPS; ++n) {    // steady state, no guards
        if (n + 8 < T_STEPS)                         // global_prefetch_b8
            __builtin_prefetch(&x[(n + 8) * B_TOTAL + batch_base + m], 0, 0);
        pipeline_iter(n, false, false);
    }
#pragma unroll 1
    for (int n = T_STEPS; n < T_STEPS + NLAYERS - 1; ++n)  // ramp-down
        pipeline_iter(n, false, true);

    // ---- Decoder on last hidden of top layer --------------------------------
    const int jb = hi ? 2 : 0;
    float part = hst[NLAYERS - 1][0] * w_dec[jb] +
                 hst[NLAYERS - 1][1] * w_dec[jb + 1];
    float y = part + half_swap(part) + b_dec[0];     // full EXEC before branch
    if (!hi) out[batch_base + m] = y;
}

extern "C" void kernel_launch(void* const* d_in, const int* in_sizes, int n_in,
                              void* d_out, int out_size, void* d_ws, size_t ws_size,
                              hipStream_t stream) {
    (void)in_sizes; (void)n_in; (void)out_size; (void)d_ws; (void)ws_size;
    const float* x         = (const float*)d_in[0];
    const float* w_ih0     = (const float*)d_in[1];
    const float* w_ih_rest = (const float*)d_in[2];
    const float* w_hh      = (const float*)d_in[3];
    const float* b_ih      = (const float*)d_in[4];
    const float* b_hh      = (const float*)d_in[5];
    const float* w_dec     = (const float*)d_in[6];
    const float* b_dec     = (const float*)d_in[7];
    float* out             = (float*)d_out;

    // 16 batch / wave, 4 waves / block -> 64 blocks x 128 threads covers B=4096
    dim3 grid(B_TOTAL / (16 * 4));
    dim3 block(128);
    lstm10_fused_kernel<<<grid, block, 0, stream>>>(
        x, w_ih0, w_ih_rest, w_hh, b_ih, b_hh, w_dec, b_dec, out);
}